// SOENModel_38319698215674
// MI455X (gfx1250) — compile-verified
//
#include <hip/hip_runtime.h>

#define HID    64
#define TILE_C 32          // columns per block
#define TILE_R 128         // rows per LDS tile
#define LSTRIDE (TILE_R + 1)   // pad: stride 129 floats -> conflict-free both ways

typedef __attribute__((ext_vector_type(2))) float v2f;
typedef __attribute__((ext_vector_type(8))) float v8f;

#if defined(__has_builtin)
#  if __has_builtin(__builtin_amdgcn_wmma_f32_16x16x4_f32)
#    define USE_WMMA_F32K4 1
#  endif
#  if __has_builtin(__builtin_amdgcn_tanhf)
#    define USE_HW_TANH 1
#  endif
#endif

__device__ __forceinline__ float fast_tanh(float x) {
#ifdef USE_HW_TANH
  return __builtin_amdgcn_tanhf(x);     // gfx1250 v_tanh_f32
#else
  float e = __expf(2.0f * x);
  return 1.0f - 2.0f / (e + 1.0f);
#endif
}

__global__ __launch_bounds__(256) void soen_fused(
    const float* __restrict__ phi, const float* __restrict__ s,
    const float* __restrict__ i_b, const float* __restrict__ w1,
    const float* __restrict__ b1,  const float* __restrict__ w2,
    const float* __restrict__ b2,  const int* __restrict__ func_idx,
    float* __restrict__ out, int N, int Brows)
{
  __shared__ float lp[TILE_C * LSTRIDE];   // p tile (overwritten in place with g)
  __shared__ float ls[TILE_C * LSTRIDE];   // s tile
  __shared__ float s_ib[TILE_C];
  __shared__ int   s_fi[TILE_C];

  const int tid  = threadIdx.x;
  const int lane = tid & 31;               // wave32
  const int wave = tid >> 5;               // 8 waves / block
  const int n0   = blockIdx.x * TILE_C;
  const int hl   = lane & 15;

  if (tid < TILE_C) {
    s_fi[tid] = func_idx[n0 + tid];
    s_ib[tid] = i_b[n0 + tid];
  }

  const float b2v = b2[0];

#ifdef USE_WMMA_F32K4
  // Per-lane constant operands for V_WMMA_F32_16X16X4_F32.
  // B (4x16, K x N) tile t: lanes 0-15 hold {K0,K1} of column N=hl,
  // lanes 16-31 hold {K2,K3} of column N=hl (K3 is the zero pad).
  v2f   bt[4];
  float w2v[4], b1v[4];
#pragma unroll
  for (int t = 0; t < 4; ++t) {
    const int h = 16 * t + hl;
    if (lane < 16) { bt[t].x = w1[0 * HID + h]; bt[t].y = w1[1 * HID + h]; }
    else           { bt[t].x = w1[2 * HID + h]; bt[t].y = 0.0f; }
    w2v[t] = w2[h];
    b1v[t] = b1[h];
  }
#endif

  const int nRT = Brows / TILE_R;
  for (int rt = 0; rt < nRT; ++rt) {
    const int rbase = rt * TILE_R;

    // ---- load phase: coalesced global -> LDS transpose (lane = column) ----
    for (int r = wave * (TILE_R / 8); r < (wave + 1) * (TILE_R / 8); ++r) {
      const size_t gidx = (size_t)(rbase + r) * (size_t)N + (size_t)(n0 + lane);
      float pv = phi[gidx];
      pv = pv - rintf(pv);                  // periodic(): round-half-even
      lp[lane * LSTRIDE + r] = pv;
      ls[lane * LSTRIDE + r] = s[gidx];
    }
    __syncthreads();

    // ---- compute phase: one wave per column -> func_idx is wave-uniform ----
    for (int j = 0; j < 4; ++j) {
      const int   c   = wave + 8 * j;
      const int   fi  = __builtin_amdgcn_readfirstlane(s_fi[c]);
      const float ibv = s_ib[c];
      float*       colp = &lp[c * LSTRIDE];
      const float* cols = &ls[c * LSTRIDE];

      if (fi == 3) {
#ifdef USE_WMMA_F32K4
        // MLP via WMMA: 16 elements (rows) per step, full wave active.
        for (int m = 0; m < TILE_R; m += 16) {
          // A (16x4): lanes 0-15 = {p,s} (K0,K1) of row m+lane;
          //           lanes 16-31 = {ib,0} (K2,K3).
          v2f a;
          if (lane < 16) { a.x = colp[m + lane]; a.y = cols[m + lane]; }
          else           { a.x = ibv;            a.y = 0.0f; }

          float part[8] = {0.f, 0.f, 0.f, 0.f, 0.f, 0.f, 0.f, 0.f};
#pragma unroll
          for (int t = 0; t < 4; ++t) {
            v8f acc = {b1v[t], b1v[t], b1v[t], b1v[t],
                       b1v[t], b1v[t], b1v[t], b1v[t]};   // bias folded into C
            acc = __builtin_amdgcn_wmma_f32_16x16x4_f32(
                false, a, false, bt[t], (short)0, acc, false, false);
#pragma unroll
            for (int r = 0; r < 8; ++r)
              part[r] += fast_tanh(acc[r]) * w2v[t];       // layer-2 partial dot
          }
          // Reduce over the 16 N-lanes of each half (C layout: VGPR r ->
          // M=r in lanes 0-15, M=r+8 in lanes 16-31).
#pragma unroll
          for (int r = 0; r < 8; ++r) {
            float v = part[r];
            v += __shfl_xor(v, 1, 32);
            v += __shfl_xor(v, 2, 32);
            v += __shfl_xor(v, 4, 32);
            v += __shfl_xor(v, 8, 32);
            part[r] = v + b2v;
          }
          if (lane == 0) {
#pragma unroll
            for (int r = 0; r < 8; ++r) colp[m + r] = part[r];
          } else if (lane == 16) {
#pragma unroll
            for (int r = 0; r < 8; ++r) colp[m + 8 + r] = part[r];
          }
        }
#else
        // VALU fallback MLP (32 rows per step)
        for (int m = 0; m < TILE_R; m += 32) {
          const float p  = colp[m + lane];
          const float sv = cols[m + lane];
          float acc = b2v;
          for (int h = 0; h < HID; ++h) {
            float pre = fmaf(p, w1[h],
                        fmaf(sv, w1[HID + h],
                        fmaf(ibv, w1[2 * HID + h], b1[h])));
            acc = fmaf(fast_tanh(pre), w2[h], acc);
          }
          colp[m + lane] = acc;
        }
#endif
      } else {
        // cheap branches: 32 rows per step, no divergence (fi uniform)
        for (int m = 0; m < TILE_R; m += 32) {
          const float p  = colp[m + lane];
          const float sv = cols[m + lane];
          float g;
          if (fi == 0)      g = fmaxf(fabsf(p) - sv, 0.0f);
          else if (fi == 1) g = fast_tanh(p) * (1.0f - sv);
          else              g = __expf(-10.0f * p * p) - sv;  // exp(-p^2/0.1)-s
          colp[m + lane] = g;
        }
      }
    }
    __syncthreads();

    // ---- store phase: coalesced LDS -> global ----
    for (int r = wave * (TILE_R / 8); r < (wave + 1) * (TILE_R / 8); ++r) {
      out[(size_t)(rbase + r) * (size_t)N + (size_t)(n0 + lane)] =
          lp[lane * LSTRIDE + r];
    }
    __syncthreads();   // LDS reused next row-tile
  }
}

extern "C" void kernel_launch(void* const* d_in, const int* in_sizes, int n_in,
                              void* d_out, int out_size, void* d_ws, size_t ws_size,
                              hipStream_t stream) {
  (void)n_in; (void)d_ws; (void)ws_size; (void)out_size;
  const float* phi = (const float*)d_in[0];
  const float* s   = (const float*)d_in[1];
  const float* ib  = (const float*)d_in[2];
  const float* w1  = (const float*)d_in[3];
  const float* b1  = (const float*)d_in[4];
  const float* w2  = (const float*)d_in[5];
  const float* b2  = (const float*)d_in[6];
  const int*   fi  = (const int*)d_in[7];
  float* out = (float*)d_out;

  const int N = in_sizes[2];          // 65536 (i_b length)
  const int B = in_sizes[0] / N;      // 256

  dim3 grid(N / TILE_C);
  dim3 block(256);
  soen_fused<<<grid, block, 0, stream>>>(phi, s, ib, w1, b1, w2, b2, fi,
                                         out, N, B);
}